// LSTM_torch_73538430042122
// MI455X (gfx1250) — compile-verified
//
#include <hip/hip_runtime.h>
#include <math.h>
#include <stdint.h>

// Problem dims
#define V_  10000
#define E_  256
#define H_  512
#define B_  8
#define S_  512
#define GH  2048     // 4*H

typedef __attribute__((ext_vector_type(2))) float v2f;
typedef __attribute__((ext_vector_type(8))) float v8f;

// D = A(16x4) * B(4x16) + C, fp32, wave32.
__device__ __forceinline__ v8f wmma4(v2f a, v2f b, v8f c) {
  return __builtin_amdgcn_wmma_f32_16x16x4_f32(
      false, a, false, b, (short)0, c, false, false);
}

// CDNA5 async DMA: memory -> LDS, 16B per lane, tracked by ASYNCcnt.
__device__ __forceinline__ void async_ld_b128(void* lds_ptr, const void* gptr) {
  unsigned lds_off = (unsigned)(uintptr_t)lds_ptr;   // generic LDS addr: low 32 = offset
  asm volatile("global_load_async_to_lds_b128 %0, %1, off"
               :: "v"(lds_off), "v"(gptr) : "memory");
}
__device__ __forceinline__ void async_wait0() {
  asm volatile("s_wait_asynccnt 0x0" ::: "memory");
}

__device__ __forceinline__ float sigm(float x) { return 1.0f / (1.0f + __expf(-x)); }

// ---------------------------------------------------------------- init
__global__ void init_ws_kernel(float* hbuf, unsigned* bar) {
  int i = blockIdx.x * blockDim.x + threadIdx.x;
  if (i < 2 * B_ * H_) hbuf[i] = 0.0f;
  if (i < 4) bar[i] = 0u;
}

// ---------------------------------------------------------------- embedding gather
__global__ void embed_gather_kernel(const int* __restrict__ tokens,
                                    const float* __restrict__ embed,
                                    float* __restrict__ x) {
  int i = blockIdx.x * blockDim.x + threadIdx.x;  // < B*S*E
  int e = i & (E_ - 1);
  int r = i >> 8;
  x[i] = embed[(long)tokens[r] * E_ + e];
}

// ---------------------------------------------------------------- grid barrier
__device__ __forceinline__ void grid_barrier(unsigned* bar, unsigned nblocks, unsigned gen) {
  __syncthreads();
  if (threadIdx.x == 0) {
    unsigned prev = __hip_atomic_fetch_add(&bar[0], 1u, __ATOMIC_RELEASE,
                                           __HIP_MEMORY_SCOPE_AGENT);
    if (prev == nblocks * (gen + 1u) - 1u) {
      __hip_atomic_store(&bar[1], gen + 1u, __ATOMIC_RELEASE, __HIP_MEMORY_SCOPE_AGENT);
    } else {
      while (__hip_atomic_load(&bar[1], __ATOMIC_ACQUIRE, __HIP_MEMORY_SCOPE_AGENT) <
             gen + 1u) {
        __builtin_amdgcn_s_sleep(2);
      }
    }
  }
  __syncthreads();
}

// ---------------------------------------------------------------- LSTM (persistent, cooperative)
#define LSTM_BLOCKS  32
#define LSTM_THREADS 256   // 8 waves: 4 N-tiles x 2 K-halves
#define KTOT 768           // E + H
#define KHALF 384
#define HJ   16            // hidden units per block
#define NCOLS 64           // 4 gates * HJ
#define ACT_STRIDE 772     // 768 + 4 pad (row pitch 3088 B, 16B-aligned)

__global__ __launch_bounds__(LSTM_THREADS)
void lstm_kernel(const float* __restrict__ x, const float* __restrict__ w_ih,
                 const float* __restrict__ w_hh, const float* __restrict__ bias,
                 float* __restrict__ hs, float* hbuf, unsigned* bar,
                 float* __restrict__ out_tail) {
  extern __shared__ float smem[];
  float* w_lds = smem;                          // [KTOT][NCOLS]  192 KB
  float* act   = smem + KTOT * NCOLS;           // [9][ACT_STRIDE] (row 8 = zeros)
  float* glp   = act + 9 * ACT_STRIDE;          // [2 khalf][8 b][64 c]

  const int g    = blockIdx.x;
  const int tid  = threadIdx.x;
  const int lane = tid & 31;
  const int wv   = tid >> 5;                    // 0..7
  const int tile = wv & 3;                      // N-tile
  const int kh   = wv >> 2;                     // K half

  // One-time weight-slice preload via async DMA (12288 b128 chunks).
  // Local col c -> global col (c/16)*H + g*HJ + (c%16); 4-col chunks stay inside a gate group.
  for (int q = tid; q < (KTOT * NCOLS) / 4; q += LSTM_THREADS) {
    int k = q >> 4, c = (q & 15) << 2;
    int gc = (c >> 4) * H_ + g * HJ + (c & 15);
    const float* src = (k < E_) ? (w_ih + (long)k * GH + gc)
                                : (w_hh + (long)(k - E_) * GH + gc);
    async_ld_b128(&w_lds[k * NCOLS + c], src);
  }
  // Zero pad row (A-fragment rows M>=8) — keeps the hot loop divergence-free.
  for (int idx = tid; idx < ACT_STRIDE; idx += LSTM_THREADS) act[8 * ACT_STRIDE + idx] = 0.0f;

  const int b_own = (tid & 127) >> 4;           // 0..7
  const int j_own = tid & 15;                   // 0..15
  const int jg    = g * HJ + j_own;
  // Bias is loop-invariant: load once.
  const float bias_i = bias[         jg];
  const float bias_f = bias[H_     + jg];
  const float bias_g = bias[2 * H_ + jg];
  const float bias_o = bias[3 * H_ + jg];
  float c_state = 0.0f;

  const int koff = (lane < 16) ? 0 : 2;
  const int ml   = lane & 15;
  const int mrow = (ml < 8) ? ml : 8;           // pad row for M>=8
  const float* arow = act + mrow * ACT_STRIDE;
  const int n0c  = tile * 16 + ml;
  const int kbeg = kh * KHALF + koff;
  const int kend = kh * KHALF + KHALF;

  for (int t = 0; t < S_; ++t) {
    // Stage x_t via async DMA (512 b128 chunks), h_{t-1} via device-coherent loads.
    for (int q = tid; q < (B_ * E_) / 4; q += LSTM_THREADS) {   // 2 per thread
      int b = q >> 6, c4 = (q & 63) << 2;
      async_ld_b128(&act[b * ACT_STRIDE + c4], x + ((long)b * S_ + t) * E_ + c4);
    }
    volatile const float* hprev = hbuf + (t & 1) * (B_ * H_);
    for (int idx = tid; idx < B_ * H_; idx += LSTM_THREADS) {   // 16 per thread
      int b = idx >> 9, k = idx & 511;
      act[b * ACT_STRIDE + E_ + k] = hprev[b * H_ + k];
    }
    async_wait0();
    __syncthreads();

    // Partial gates over this wave's K half; 1-deep software pipeline.
    v8f acc = {0.f, 0.f, 0.f, 0.f, 0.f, 0.f, 0.f, 0.f};
    v2f a0, b0;
    a0.x = arow[kbeg];                 a0.y = arow[kbeg + 1];
    b0.x = w_lds[kbeg * NCOLS + n0c];  b0.y = w_lds[(kbeg + 1) * NCOLS + n0c];
#pragma unroll 2
    for (int k = kbeg + 4; k < kend; k += 4) {
      v2f a1, b1;
      a1.x = arow[k];                  a1.y = arow[k + 1];
      b1.x = w_lds[k * NCOLS + n0c];   b1.y = w_lds[(k + 1) * NCOLS + n0c];
      acc = wmma4(a0, b0, acc);
      a0 = a1; b0 = b1;
    }
    acc = wmma4(a0, b0, acc);

    // Write partial C (valid rows M=0..7: lanes 0-15, VGPRs 0..7).
    if (lane < 16) {
#pragma unroll
      for (int r = 0; r < 8; ++r) glp[kh * 512 + r * NCOLS + tile * 16 + lane] = acc[r];
    }
    __syncthreads();

    // Pointwise cell update: threads 0..127 = 8 batch x 16 hidden.
    if (tid < B_ * HJ) {
      float iv = glp[b_own * NCOLS +      j_own] + glp[512 + b_own * NCOLS +      j_own] + bias_i;
      float fv = glp[b_own * NCOLS + 16 + j_own] + glp[512 + b_own * NCOLS + 16 + j_own] + bias_f;
      float gv = glp[b_own * NCOLS + 32 + j_own] + glp[512 + b_own * NCOLS + 32 + j_own] + bias_g;
      float ov = glp[b_own * NCOLS + 48 + j_own] + glp[512 + b_own * NCOLS + 48 + j_own] + bias_o;
      c_state = sigm(fv) * c_state + sigm(iv) * tanhf(gv);
      float hv = sigm(ov) * tanhf(c_state);

      hbuf[((t + 1) & 1) * (B_ * H_) + b_own * H_ + jg] = hv;
      hs[((long)b_own * S_ + t) * H_ + jg] = hv;
      if (t == S_ - 1) {
        out_tail[b_own * H_ + jg] = hv;                 // h_T
        out_tail[B_ * H_ + b_own * H_ + jg] = c_state;  // c_T
      }
    }
    grid_barrier(bar, LSTM_BLOCKS, (unsigned)t);
  }
}

// ---------------------------------------------------------------- output projection (WMMA GEMM)
// Block = 64 rows (4 M-tiles) x 128 cols (8 waves x 1 N-tile); one B-fragment
// feeds 4 WMMAs. A tile staged by async DMA.
#define PROJ_THREADS 256
#define MREP 4
#define LDSTRIDE 516       // pitch 2064 B, 16B-aligned

__global__ __launch_bounds__(PROJ_THREADS)
void proj_kernel(const float* __restrict__ hs, const float* __restrict__ w_out,
                 const float* __restrict__ b_out, float* __restrict__ out) {
  extern __shared__ float a_lds[];              // [64][LDSTRIDE] = 129 KB
  const int row0 = blockIdx.y * (MREP * 16);    // 64 blocks * 64 rows = 4096
  const int tid  = threadIdx.x;

  for (int q = tid; q < (MREP * 16 * H_) / 4; q += PROJ_THREADS) {  // 32 per thread
    int r = q >> 7, c4 = (q & 127) << 2;
    async_ld_b128(&a_lds[r * LDSTRIDE + c4], hs + (long)(row0 + r) * H_ + c4);
  }
  async_wait0();
  __syncthreads();

  const int wv = tid >> 5, lane = tid & 31;
  const int ntile = blockIdx.x * 8 + wv;        // 625 N-tiles
  if (ntile >= 625) return;                     // wave-uniform
  const int n0   = ntile * 16;
  const int koff = (lane < 16) ? 0 : 2;
  const int ml   = lane & 15;

  v8f acc[MREP];
#pragma unroll
  for (int i = 0; i < MREP; ++i) acc[i] = (v8f){0.f, 0.f, 0.f, 0.f, 0.f, 0.f, 0.f, 0.f};

  const float* bp = w_out + n0 + ml;
#pragma unroll 2
  for (int k = 0; k < H_; k += 4) {
    v2f b;
    b.x = bp[(long)(k + koff) * V_];
    b.y = bp[(long)(k + koff + 1) * V_];
#pragma unroll
    for (int i = 0; i < MREP; ++i) {
      const float* ap = &a_lds[(i * 16 + ml) * LDSTRIDE + k + koff];
      v2f a; a.x = ap[0]; a.y = ap[1];
      acc[i] = wmma4(a, b, acc[i]);
    }
  }

  const float bo = b_out[n0 + ml];
#pragma unroll
  for (int i = 0; i < MREP; ++i) {
    const int mbase = row0 + i * 16 + ((lane < 16) ? 0 : 8);
#pragma unroll
    for (int r = 0; r < 8; ++r) {
      float v = acc[i][r] + bo;
      v = v > 0.0f ? v : 0.0f;                  // relu
      out[(long)(mbase + r) * V_ + n0 + ml] = v;
    }
  }
}

// ---------------------------------------------------------------- log_softmax over axis 0 (B=8), in place
__global__ void logsoftmax_kernel(float* __restrict__ out) {
  long i = (long)blockIdx.x * blockDim.x + threadIdx.x;   // < S*V
  const long stride = (long)S_ * V_;
  float z[B_];
  float m = -3.4e38f;
#pragma unroll
  for (int b = 0; b < B_; ++b) { z[b] = out[b * stride + i]; m = fmaxf(m, z[b]); }
  float s = 0.0f;
#pragma unroll
  for (int b = 0; b < B_; ++b) s += __expf(z[b] - m);
  float lse = m + __logf(s);
#pragma unroll
  for (int b = 0; b < B_; ++b)
    __builtin_nontemporal_store(z[b] - lse, &out[b * stride + i]);  // final, bypass L2 rinse
}

// ---------------------------------------------------------------- launch
extern "C" void kernel_launch(void* const* d_in, const int* in_sizes, int n_in,
                              void* d_out, int out_size, void* d_ws, size_t ws_size,
                              hipStream_t stream) {
  const int*   tokens = (const int*)  d_in[0];
  const float* embed  = (const float*)d_in[1];
  const float* w_ih   = (const float*)d_in[2];
  const float* w_hh   = (const float*)d_in[3];
  const float* bias   = (const float*)d_in[4];
  const float* w_out  = (const float*)d_in[5];
  const float* b_out  = (const float*)d_in[6];
  float* out = (float*)d_out;

  // Workspace (floats): x | hs | hbuf(2xBxH) | barrier  (~12.6 MB)
  float* ws   = (float*)d_ws;
  float* x    = ws;
  float* hs   = x + (size_t)B_ * S_ * E_;
  float* hbuf = hs + (size_t)B_ * S_ * H_;
  unsigned* bar = (unsigned*)(hbuf + 2 * B_ * H_);

  init_ws_kernel<<<32, 256, 0, stream>>>(hbuf, bar);
  embed_gather_kernel<<<(B_ * S_ * E_) / 256, 256, 0, stream>>>(tokens, embed, x);

  size_t lstm_lds = (size_t)(KTOT * NCOLS + 9 * ACT_STRIDE + 2 * 8 * NCOLS) * sizeof(float);
  lstm_kernel<<<LSTM_BLOCKS, LSTM_THREADS, lstm_lds, stream>>>(
      x, w_ih, w_hh, bias, hs, hbuf, bar, out + (size_t)B_ * S_ * V_);

  size_t proj_lds = (size_t)(MREP * 16 * LDSTRIDE) * sizeof(float);
  proj_kernel<<<dim3(79, 64), PROJ_THREADS, proj_lds, stream>>>(hs, w_out, b_out, out);

  logsoftmax_kernel<<<(S_ * V_) / 256, 256, 0, stream>>>(out);
}